// MyAttention_Multihead_projection_6459630814001
// MI455X (gfx1250) — compile-verified
//
#include <hip/hip_runtime.h>

// B=16, N1=32, NL=256, D_MODEL=16, A=2, D_K=8, D_V=8, D_OUT=16
// One workgroup per (b,n): 512 WGs x 256 threads (8 wave32).
// Fully fused: projections -> attn (WMMA) -> relu -> mean out (NT stores) ->
// attn@Vp (WMMA, both heads packed in one B operand) -> Wo (WMMA).

typedef __attribute__((ext_vector_type(2))) float v2f;
typedef __attribute__((ext_vector_type(8))) float v8f;

static __device__ __forceinline__ v8f wmma_f32(v2f a, v2f b, v8f c) {
    // (neg_a, A, neg_b, B, c_mod, C, reuse_a, reuse_b)
    return __builtin_amdgcn_wmma_f32_16x16x4_f32(false, a, false, b, (short)0, c, false, false);
}

// single v_max_num_f32 with inline 0 (NaN -> 0, -0 -> +0; matches jax relu on finite data)
static __device__ __forceinline__ float relu1(float x) { return x > 0.f ? x : 0.f; }

__global__ __launch_bounds__(256)
void attn_fused_kernel(const float* __restrict__ Q, const float* __restrict__ K,
                       const float* __restrict__ V, const float* __restrict__ Wq,
                       const float* __restrict__ Wk, const float* __restrict__ Wv,
                       const float* __restrict__ Wo,
                       float* __restrict__ proj, float* __restrict__ attnm)
{
    // Projected tensors, packed column c = a*8 + j, padded stride 17 (bank-conflict free)
    __shared__ float sQp[256][17];
    __shared__ float sKp[256][17];
    __shared__ float sVp[256][17];
    __shared__ float swq[256];
    __shared__ float swk[256];
    __shared__ float swv[256];
    __shared__ float swo[16][17];      // swo[c][z] = Wo[m=c&7][a=c>>3][z]
    __shared__ float ssc[8][16][17];   // per-wave D->A bounce scratch

    const int t  = threadIdx.x;
    const int bn = blockIdx.x;

    // ---- stage weights (hot across all 512 WGs: keep default RT caching) ----
    swq[t] = Wq[t];
    swk[t] = Wk[t];
    swv[t] = Wv[t];
    {
        const int c = t >> 4, z = t & 15;
        swo[c][z] = Wo[(c & 7) * 32 + (c >> 3) * 16 + z];
    }
    __syncthreads();

    // ---- projections: thread t owns row t (coalesced 64B row loads, read-once -> NT) ----
    {
        const float* qr = Q + ((long)bn * 256 + t) * 16;
        const float* kr = K + ((long)bn * 256 + t) * 16;
        const float* vr = V + ((long)bn * 256 + t) * 16;
        float q[16], k[16], v[16];
#pragma unroll
        for (int i = 0; i < 16; ++i) {
            q[i] = __builtin_nontemporal_load(qr + i);
            k[i] = __builtin_nontemporal_load(kr + i);
            v[i] = __builtin_nontemporal_load(vr + i);
        }
#pragma unroll
        for (int c = 0; c < 16; ++c) {
            float aq = 0.f, ak = 0.f, av = 0.f;
#pragma unroll
            for (int i = 0; i < 16; ++i) {
                aq = fmaf(q[i], swq[i * 16 + c], aq);
                ak = fmaf(k[i], swk[i * 16 + c], ak);
                av = fmaf(v[i], swv[i * 16 + c], av);
            }
            sQp[t][c] = aq;
            sKp[t][c] = ak;
            sVp[t][c] = av;
        }
    }
    __syncthreads();

    const int lane = t & 31;
    const int wave = t >> 5;
    const int lh   = lane >> 4;   // half-wave: K-offset (+2) in A/B frags, M-offset (+8) in C/D
    const int lm   = lane & 15;   // M (A) or N (B/C/D) within tile
    float (*sc)[17] = ssc[wave];

    for (int ii = 0; ii < 2; ++ii) {
        const int k0 = (wave * 2 + ii) * 16;   // row-tile base

        // A-fragments of Qp for attn (per head, K=8 -> 2 K-steps of 4), z-invariant
        v2f aQ[2][2];
#pragma unroll
        for (int h = 0; h < 2; ++h)
#pragma unroll
            for (int kk = 0; kk < 2; ++kk) {
                const int j = h * 8 + kk * 4 + lh * 2;
                aQ[h][kk].x = sQp[k0 + lm][j];
                aQ[h][kk].y = sQp[k0 + lm][j + 1];
            }

        v8f acc0 = {0.f,0.f,0.f,0.f,0.f,0.f,0.f,0.f};   // emb accum, head0-valid cols 0..7
        v8f acc1 = {0.f,0.f,0.f,0.f,0.f,0.f,0.f,0.f};   // emb accum, head1-valid cols 8..15

        for (int zt = 0; zt < 16; ++zt) {
            const int z0 = zt * 16;

            // B-fragments: B[j][z] = Kp[z0+z][j]
            v2f bK[2][2];
#pragma unroll
            for (int h = 0; h < 2; ++h)
#pragma unroll
                for (int kk = 0; kk < 2; ++kk) {
                    const int j = h * 8 + kk * 4 + lh * 2;
                    bK[h][kk].x = sKp[z0 + lm][j];
                    bK[h][kk].y = sKp[z0 + lm][j + 1];
                }

            v8f att0 = {0.f,0.f,0.f,0.f,0.f,0.f,0.f,0.f};
            v8f att1 = {0.f,0.f,0.f,0.f,0.f,0.f,0.f,0.f};
            att0 = wmma_f32(aQ[0][0], bK[0][0], att0);
            att0 = wmma_f32(aQ[0][1], bK[0][1], att0);
            att1 = wmma_f32(aQ[1][0], bK[1][0], att1);
            att1 = wmma_f32(aQ[1][1], bK[1][1], att1);

#pragma unroll
            for (int r = 0; r < 8; ++r) {
                att0[r] = relu1(att0[r]);
                att1[r] = relu1(att1[r]);
            }

            // attn_mean (head average) straight from D layout: coalesced, streaming (NT)
            {
                float* ab = attnm + ((long)bn * 256 + k0 + 8 * lh) * 256 + z0 + lm;
#pragma unroll
                for (int r = 0; r < 8; ++r)
                    __builtin_nontemporal_store(0.5f * (att0[r] + att1[r]), ab + (long)r * 256);
            }

            // D->A bounce head0 through wave-private scratch.
            // LDS ops from one wave execute in order; compiler inserts minimal dscnt waits.
#pragma unroll
            for (int r = 0; r < 8; ++r) sc[r + 8 * lh][lm] = att0[r];
            asm volatile("" ::: "memory");
            v2f aA0[4];
#pragma unroll
            for (int kk = 0; kk < 4; ++kk) {
                const int j = kk * 4 + lh * 2;
                aA0[kk].x = sc[lm][j];
                aA0[kk].y = sc[lm][j + 1];
            }
            asm volatile("" ::: "memory");
            // D->A bounce head1 (in-order LDS: WAR on scratch is safe)
#pragma unroll
            for (int r = 0; r < 8; ++r) sc[r + 8 * lh][lm] = att1[r];
            asm volatile("" ::: "memory");
            v2f aA1[4];
#pragma unroll
            for (int kk = 0; kk < 4; ++kk) {
                const int j = kk * 4 + lh * 2;
                aA1[kk].x = sc[lm][j];
                aA1[kk].y = sc[lm][j + 1];
            }

            // emb += attn_h @ VpPacked ; B holds BOTH heads' V columns (c = a*8+m),
            // shared between the two heads' WMMAs; wrong-head columns discarded later.
#pragma unroll
            for (int kk = 0; kk < 4; ++kk) {
                const int zr = z0 + kk * 4 + lh * 2;
                v2f bV;
                bV.x = sVp[zr][lm];
                bV.y = sVp[zr + 1][lm];
                acc0 = wmma_f32(aA0[kk], bV, acc0);
                acc1 = wmma_f32(aA1[kk], bV, acc1);
            }
        }

        // Select valid head half per lane column (c = lm): c<8 -> head0, else head1
        v8f e;
#pragma unroll
        for (int r = 0; r < 8; ++r) e[r] = (lm < 8) ? acc0[r] : acc1[r];

        // D->A bounce emb, then proj = embPacked(16x16) @ WoFlat(16x16)
#pragma unroll
        for (int r = 0; r < 8; ++r) sc[r + 8 * lh][lm] = e[r];
        asm volatile("" ::: "memory");

        v8f accP = {0.f,0.f,0.f,0.f,0.f,0.f,0.f,0.f};
#pragma unroll
        for (int kk = 0; kk < 4; ++kk) {
            const int j = kk * 4 + lh * 2;
            v2f aE, bW;
            aE.x = sc[lm][j];
            aE.y = sc[lm][j + 1];
            bW.x = swo[j][lm];
            bW.y = swo[j + 1][lm];
            accP = wmma_f32(aE, bW, accP);
        }

        float* pb = proj + ((long)bn * 256 + k0 + 8 * lh) * 16 + lm;
#pragma unroll
        for (int r = 0; r < 8; ++r) __builtin_nontemporal_store(accP[r], pb + (long)r * 16);
    }
}

extern "C" void kernel_launch(void* const* d_in, const int* in_sizes, int n_in,
                              void* d_out, int out_size, void* d_ws, size_t ws_size,
                              hipStream_t stream) {
    const float* Q  = (const float*)d_in[0];
    const float* K  = (const float*)d_in[1];
    const float* V  = (const float*)d_in[2];
    const float* Wq = (const float*)d_in[3];
    const float* Wk = (const float*)d_in[4];
    const float* Wv = (const float*)d_in[5];
    const float* Wo = (const float*)d_in[6];

    float* proj  = (float*)d_out;                                  // [16,32,256,16]
    float* attnm = (float*)d_out + (size_t)16 * 32 * 256 * 16;     // [16,32,256,256]

    attn_fused_kernel<<<dim3(16 * 32), dim3(256), 0, stream>>>(
        Q, K, V, Wq, Wk, Wv, Wo, proj, attnm);
}